// Neural_time_50337016709696
// MI455X (gfx1250) — compile-verified
//
#include <hip/hip_runtime.h>

// Problem constants (from reference)
#define NMOD   3
#define NVEC   500000
#define RDIM   32
#define NFF    256
#define BATCH  131072
#define IN_DIM 97                 // R*NMOD + 1
#define NTILES (BATCH / 16)       // 8192 tiles of 16 batch rows

// LDS layout strides (dwords) -- chosen for conflict-free ds_load_b64
#define WT_STRIDE 102             // W^T rows (per output column), 97 used + pad
#define IN_STRIDE 100             // per-batch-row input vector, 97 used + pad
#define WAVES_PER_BLOCK 8
#define NBLOCKS 256

typedef __attribute__((ext_vector_type(2))) float v2f;
typedef __attribute__((ext_vector_type(8))) float v8f;

__global__ __launch_bounds__(256) void rff_fused_wmma_f32(
    const int*   __restrict__ b_i_n,   // [B, NMOD]
    const float* __restrict__ b_t_n,   // [B]
    const float* __restrict__ U,       // [NMOD, NVEC, RDIM]
    const float* __restrict__ W_freq,  // [IN_DIM, NFF]
    const float* __restrict__ w_out,   // [2*NFF]
    const float* __restrict__ b_out,   // [1]
    float*       __restrict__ out)     // [B]
{
    __shared__ float wtLds[NFF * WT_STRIDE];                 // 104448 B: W^T [col][k]
    __shared__ float inLds[WAVES_PER_BLOCK * 16 * IN_STRIDE];// 51200 B: gathered inputs
    __shared__ float woLds[2 * NFF];                         // 2048 B: readout weights

    const int tid  = threadIdx.x;
    const int lane = tid & 31;
    const int wave = tid >> 5;

    // ---- Stage W_freq transposed into LDS: wtLds[n*WT_STRIDE + k] = W[k][n] ----
    // 256 threads: thread `tid` owns column n = tid; global reads fully coalesced.
    for (int k = 0; k < IN_DIM; ++k)
        wtLds[tid * WT_STRIDE + k] = W_freq[k * NFF + tid];
    woLds[tid]       = w_out[tid];
    woLds[NFF + tid] = w_out[NFF + tid];
    __syncthreads();

    const int rowA  = lane & 15;          // M row (A) / N col (B,C,D) for this lane
    const int kap   = (lane >> 4) << 1;   // half-wave K offset: 0 or 2
    const int hbase = (lane >> 4) << 3;   // half-wave M base for C/D: 0 or 8
    const float bo  = b_out[0];

    float* inW = inLds + wave * (16 * IN_STRIDE);
    const float* inRow = inW + rowA * IN_STRIDE;

    const int waveGlobal = blockIdx.x * WAVES_PER_BLOCK + wave;
    const int waveStride = gridDim.x * WAVES_PER_BLOCK;

    for (int tile = waveGlobal; tile < NTILES; tile += waveStride) {
        // ---- Gather: 16 rows x (3 modes x 32 floats) into this wave's LDS ----
        // 48 (row,mode) tasks over 32 lanes, each task copies 32 floats (8x b128).
        #pragma unroll
        for (int t2 = 0; t2 < 2; ++t2) {
            int task = t2 * 32 + lane;
            if (task < 48) {
                int row = task / 3;
                int m   = task - row * 3;
                int gi  = b_i_n[(tile * 16 + row) * NMOD + m];
                const float4* src =
                    (const float4*)(U + ((size_t)m * NVEC + (size_t)gi) * RDIM);
                float4* dst = (float4*)(inW + row * IN_STRIDE + m * RDIM);
                #pragma unroll
                for (int j = 0; j < 8; ++j) dst[j] = src[j];
            }
        }
        if (lane < 16) inW[lane * IN_STRIDE + 96] = b_t_n[tile * 16 + lane];

        // Same-wave DS write -> DS read: DS pipe is in-order; keep the compiler
        // from reordering and drain outstanding LDS stores.
        asm volatile("s_wait_dscnt 0" ::: "memory");

        // t values for the 8 M-rows this half-wave accumulates (LDS broadcast)
        float tv[8];
        #pragma unroll
        for (int i = 0; i < 8; ++i) tv[i] = inW[(hbase + i) * IN_STRIDE + 96];

        float acc[8];
        #pragma unroll
        for (int i = 0; i < 8; ++i) acc[i] = 0.0f;

        // ---- 16 x 256 projection via 16 N-tiles of WMMA f32 16x16x4 ----
        for (int nt = 0; nt < 16; ++nt) {
            const int n0 = nt * 16;
            const float* wtRow = wtLds + (n0 + rowA) * WT_STRIDE; // column n0+rowA

            // Fold the 97th input (t) into the accumulator: C = t ⊗ W[96,:]
            const float w96 = wtRow[96];
            v8f c;
            #pragma unroll
            for (int i = 0; i < 8; ++i) c[i] = tv[i] * w96;

            #pragma unroll
            for (int k0 = 0; k0 < 96; k0 += 4) {
                v2f a = *(const v2f*)(inRow + k0 + kap); // A[M=rowA][k0+kap .. +1]
                v2f b = *(const v2f*)(wtRow + k0 + kap); // B[k0+kap .. +1][N=n0+rowA]
                c = __builtin_amdgcn_wmma_f32_16x16x4_f32(
                        /*neg_a=*/false, a, /*neg_b=*/false, b,
                        /*c_mod=*/(short)0, c, /*reuse_a=*/false, /*reuse_b=*/false);
            }

            const float wsin = woLds[n0 + rowA];
            const float wcos = woLds[NFF + n0 + rowA];
            #pragma unroll
            for (int i = 0; i < 8; ++i)
                acc[i] += __sinf(c[i]) * wsin + __cosf(c[i]) * wcos;
        }

        // ---- Reduce across the 16 lanes of each half-wave (columns) ----
        #pragma unroll
        for (int i = 0; i < 8; ++i) {
            float v = acc[i];
            v += __shfl_xor(v, 8, 16);
            v += __shfl_xor(v, 4, 16);
            v += __shfl_xor(v, 2, 16);
            v += __shfl_xor(v, 1, 16);
            acc[i] = v;
        }

        if (rowA == 0) {
            #pragma unroll
            for (int i = 0; i < 8; ++i)
                out[tile * 16 + hbase + i] = acc[i] * (1.0f / 16.0f) + bo;
        }
    }
}

extern "C" void kernel_launch(void* const* d_in, const int* in_sizes, int n_in,
                              void* d_out, int out_size, void* d_ws, size_t ws_size,
                              hipStream_t stream) {
    (void)in_sizes; (void)n_in; (void)out_size; (void)d_ws; (void)ws_size;
    const int*   b_i_n  = (const int*)  d_in[0];
    const float* b_t_n  = (const float*)d_in[1];
    const float* U      = (const float*)d_in[2];
    const float* W_freq = (const float*)d_in[3];
    const float* w_out  = (const float*)d_in[4];
    const float* b_out  = (const float*)d_in[5];
    float* out = (float*)d_out;

    dim3 grid(NBLOCKS), block(256);
    hipLaunchKernelGGL(rff_fused_wmma_f32, grid, block, 0, stream,
                       b_i_n, b_t_n, U, W_freq, w_out, b_out, out);
}